// ClusteringLayer_ver2_81570018885867
// MI455X (gfx1250) — compile-verified
//
#include <hip/hip_runtime.h>

typedef float v2f __attribute__((ext_vector_type(2)));
typedef float v8f __attribute__((ext_vector_type(8)));

#define FDIM   32
#define KC     8
#define CHUNKS 8   // FDIM / 4

// Persistent-wave version: each wave builds the augmented B matrix (clusters)
// once in registers, then grid-strides over 16-point tiles. Per tile:
//   t = 1 + ||x||^2 + ||c||^2 - 2 x.c   via 9 chained V_WMMA_F32_16X16X4_F32,
//   q = rcp(t), row-normalized with xor-shuffle sums and native v_rcp_f32.
__global__ __launch_bounds__(256)
void dec_soft_assign_wmma(const float* __restrict__ x,        // (Nrows, 32)
                          const float* __restrict__ clusters, // (8, 32)
                          float* __restrict__ out,            // (Nrows, 8)
                          int ntiles)
{
    const int lane = threadIdx.x & 31;
    const int wave = threadIdx.x >> 5;
    const int n    = lane & 15;   // cluster column (and row-within-tile for A)
    const int h    = lane >> 4;   // K-half selector (features 2h, 2h+1 mod 4)

    // ---------------- B matrix from clusters (built ONCE per wave) ----------
    const float mask = (n < KC) ? 1.0f : 0.0f;
    const int   crow = (n < KC) ? n : (KC - 1);  // clamp to avoid OOB reads
    v2f   braw[CHUNKS];
    float ccpart = 0.0f;
    #pragma unroll
    for (int c = 0; c < CHUNKS; ++c) {
        const v2f v = *(const v2f*)(clusters + crow * FDIM + 4 * c + 2 * h);
        ccpart += v.x * v.x + v.y * v.y;
        braw[c] = v;
    }
    // combine the two feature halves (lanes l and l^16) -> ||c_n||^2
    const float cc = ccpart + __shfl_xor(ccpart, 16, 32);

    v2f b[CHUNKS + 1];
    #pragma unroll
    for (int c = 0; c < CHUNKS; ++c) {
        b[c].x = -2.0f * mask * braw[c].x;
        b[c].y = -2.0f * mask * braw[c].y;
    }
    // augmented features 32..35:  B[32][n]=1, B[33][n]=1+||c||^2, B[34..35]=0
    b[CHUNKS].x = (h == 0) ? mask : 0.0f;
    b[CHUNKS].y = (h == 0) ? mask * (1.0f + cc) : 0.0f;

    // ---------------- grid-stride over 16-point tiles -----------------------
    const int nwaves = gridDim.x * (blockDim.x >> 5);
    for (int tile = blockIdx.x * (blockDim.x >> 5) + wave; tile < ntiles;
         tile += nwaves) {
        const long base_row = (long)tile * 16;

        // A matrix: 16 points streamed from HBM (b64 per lane per chunk)
        const float* xrow = x + (base_row + n) * FDIM;
        v2f   a[CHUNKS + 1];
        float xxpart = 0.0f;
        #pragma unroll
        for (int c = 0; c < CHUNKS; ++c) {
            const v2f v = *(const v2f*)(xrow + 4 * c + 2 * h);
            xxpart += v.x * v.x + v.y * v.y;
            a[c] = v;
        }
        const float xx = xxpart + __shfl_xor(xxpart, 16, 32); // ||x_m||^2
        a[CHUNKS].x = (h == 0) ? xx   : 0.0f;  // feature 32
        a[CHUNKS].y = (h == 0) ? 1.0f : 0.0f;  // feature 33

        // 9 chained fp32 WMMAs: t = 1 + d2
        v8f t = {};
        #pragma unroll
        for (int c = 0; c <= CHUNKS; ++c) {
            t = __builtin_amdgcn_wmma_f32_16x16x4_f32(
                    /*neg_a=*/false, a[c], /*neg_b=*/false, b[c],
                    /*c_mod=*/(short)0, t, /*reuse_a=*/false, /*reuse_b=*/false);
        }

        // q = rcp(t), normalize over the 8 cluster lanes with native rcp
        float qn[8];
        #pragma unroll
        for (int r = 0; r < 8; ++r) {
            const float q = __builtin_amdgcn_rcpf(t[r]);
            float s = q;
            s += __shfl_xor(s, 1, 32);
            s += __shfl_xor(s, 2, 32);
            s += __shfl_xor(s, 4, 32);   // sum over clusters 0..7 in each half
            qn[r] = q * __builtin_amdgcn_rcpf(s);
        }

        // D layout: VGPR r <-> point row (r + 8h); lane%16 <-> cluster n.
        if (n < KC) {
            #pragma unroll
            for (int r = 0; r < 8; ++r) {
                out[(base_row + r + 8 * h) * KC + n] = qn[r];
            }
        }
    }
}

extern "C" void kernel_launch(void* const* d_in, const int* in_sizes, int n_in,
                              void* d_out, int out_size, void* d_ws, size_t ws_size,
                              hipStream_t stream) {
    const float* x        = (const float*)d_in[0];   // (B*P, 32) flat
    const float* clusters = (const float*)d_in[1];   // (8, 32)
    float*       out      = (float*)d_out;           // (B*P, 8)

    const int nrows = in_sizes[0] / FDIM;            // 524288
    const int ntiles = (nrows + 15) / 16;            // 16 points per wave-tile
    int blocks = (ntiles + 7) / 8;                   // 8 waves per 256-thr block
    if (blocks > 1024) blocks = 1024;                // persistent waves,
                                                     // ~4 tiles per wave

    dec_soft_assign_wmma<<<blocks, 256, 0, stream>>>(x, clusters, out, ntiles);
}